// TagLoss_80238579023903
// MI455X (gfx1250) — compile-verified
//
#include <hip/hip_runtime.h>
#include <math.h>

typedef __attribute__((ext_vector_type(2))) float v2f;
typedef __attribute__((ext_vector_type(8))) float v8f;

#define NKPT 512
#define DDIM 16
#define ESTRIDE 520              // padded LDS row stride (2*520 mod 64 == 16 -> half-wave banks disjoint)
#define BLOCKS_PER_BATCH 8
#define TILES_N 32               // 512/16
#define TILES_TOTAL (TILES_N*TILES_N)

__global__ void tagloss_init(float* out) {
    if (threadIdx.x == 0) out[0] = 0.0f;
}

__global__ __launch_bounds__(256)
void tagloss_main(const float* __restrict__ pred,
                  const float* __restrict__ kpt,
                  const int*   __restrict__ vis,
                  const int*   __restrict__ tag,
                  float* __restrict__ out,
                  int B)
{
    __shared__ float Es[DDIM * ESTRIDE];   // E[d][n], padded rows
    __shared__ float sqs[NKPT];            // mean_d E^2
    __shared__ float mks[NKPT];            // visibility mask as float
    __shared__ int   tgs[NKPT];            // tags
    __shared__ float red[256];
    __shared__ float kshared;

    const int b   = blockIdx.x;
    const int tid = threadIdx.x;

    // ---------------- Phase 1: gather embeddings at keypoints into LDS ----------------
    for (int n = tid; n < NKPT; n += 256) {
        const float k0 = kpt[((size_t)b * NKPT + n) * 2 + 0];
        const float k1 = kpt[((size_t)b * NKPT + n) * 2 + 1];
        const int y = ((int)floorf(k0 * 256.0f)) & 255;   // idx[...,0] * W indexes H
        const int x = ((int)floorf(k1 * 256.0f)) & 255;   // idx[...,1] * H indexes W
        const float* pb = pred + (((size_t)b * DDIM) * 256 + (size_t)y) * 256 + (size_t)x;
        float ssum = 0.0f;
        #pragma unroll
        for (int d = 0; d < DDIM; ++d) {
            const float v = pb[(size_t)d * 256 * 256];
            Es[d * ESTRIDE + n] = v;
            ssum += v * v;
        }
        sqs[n] = ssum * (1.0f / (float)DDIM);
        mks[n] = (vis[b * NKPT + n] > 0) ? 1.0f : 0.0f;
        tgs[n] = tag[b * NKPT + n];
    }
    __syncthreads();
    if (tid == 0) {
        float k = 0.0f;
        for (int n = 0; n < NKPT; ++n) k += mks[n];
        kshared = k;
    }
    __syncthreads();

    // ---------------- Phase 2: WMMA Gram-matrix tiles + fused epilogue ----------------
    const int wave    = tid >> 5;
    const int lane    = tid & 31;
    const int laneLo  = lane & 15;
    const int halfSel = lane >> 4;

    const int tilesPerBlock = TILES_TOTAL / BLOCKS_PER_BATCH; // 128
    const int tileBase = blockIdx.y * tilesPerBlock;

    float local = 0.0f;

    for (int t = tileBase + wave; t < tileBase + tilesPerBlock; t += 8) {
        const int n0 = (t >> 5) << 4;   // row-tile origin
        const int m0 = (t & 31) << 4;   // col-tile origin

        // D[n][m] = sum_d A[n][d] * B[d][m], K=4 per WMMA, chained over D=16.
        // 32-bit A 16x4 layout: lanes 0-15 hold K={0,1}, lanes 16-31 hold K={2,3}.
        v8f c = {};
        #pragma unroll
        for (int dk = 0; dk < 4; ++dk) {
            const int dbase = dk * 4 + halfSel * 2;
            v2f a, bb;
            a.x  = Es[(dbase + 0) * ESTRIDE + n0 + laneLo];
            a.y  = Es[(dbase + 1) * ESTRIDE + n0 + laneLo];
            bb.x = Es[(dbase + 0) * ESTRIDE + m0 + laneLo];
            bb.y = Es[(dbase + 1) * ESTRIDE + m0 + laneLo];
            c = __builtin_amdgcn_wmma_f32_16x16x4_f32(
                    /*neg_a=*/false, a, /*neg_b=*/false, bb,
                    /*c_mod=*/(short)0, c, /*reuse_a=*/false, /*reuse_b=*/false);
        }

        // C/D layout: lane owns column m0+laneLo; VGPR r holds row n0 + 8*halfSel + r.
        const int   m   = m0 + laneLo;
        const float sqm = sqs[m];
        const float mkm = mks[m];
        const int   tgm = tgs[m];
        #pragma unroll
        for (int r = 0; r < 8; ++r) {
            const int   n    = n0 + halfSel * 8 + r;
            const float dotv = c[r] * (1.0f / (float)DDIM);
            const float expo = sqs[n] + sqm - 2.0f * dotv;
            // v_exp_f32 + v_rcp_f32: denominator in [1, inf) so rcp is well-conditioned
            const float psim = 2.0f * __builtin_amdgcn_rcpf(1.0f + __expf(expo));
            const float tsim = (tgs[n] == tgm) ? 1.0f : 0.0f;
            const float w    = ((tsim > 0.0f) || (psim > 0.0f)) ? 10.0f : 0.0f;
            const float dd   = psim - tsim;
            local += dd * dd * w * mks[n] * mkm;
        }
    }

    // ---------------- Phase 3: block reduction + single atomic ----------------
    red[tid] = local;
    __syncthreads();
    #pragma unroll
    for (int s = 128; s > 0; s >>= 1) {
        if (tid < s) red[tid] += red[tid + s];
        __syncthreads();
    }
    if (tid == 0) {
        const float K = kshared;
        const float denom = fmaxf(K * K, 1.0f) * (float)B;
        atomicAdd(out, red[0] / denom);
    }
}

extern "C" void kernel_launch(void* const* d_in, const int* in_sizes, int n_in,
                              void* d_out, int out_size, void* d_ws, size_t ws_size,
                              hipStream_t stream) {
    const float* pred = (const float*)d_in[0];   // [B,16,256,256] f32
    const float* kpt  = (const float*)d_in[1];   // [B,512,2] f32
    const int*   vis  = (const int*)d_in[2];     // [B,512] i32
    const int*   tag  = (const int*)d_in[3];     // [B,512] i32
    float* out = (float*)d_out;                  // scalar f32

    const int B = in_sizes[2] / NKPT;            // 32

    tagloss_init<<<1, 32, 0, stream>>>(out);
    dim3 grid(B, BLOCKS_PER_BATCH);
    tagloss_main<<<grid, 256, 0, stream>>>(pred, kpt, vis, tag, out, B);
}